// MyLossFunc_8126078124152
// MI455X (gfx1250) — compile-verified
//
#include <hip/hip_runtime.h>

#define KTOP 50
#define TPB 256
#define NB1 2048
#define CAND_MAX 256

typedef __attribute__((ext_vector_type(2))) float v2f;
typedef __attribute__((ext_vector_type(8))) float v8f;

// Order-preserving float -> uint key (larger float => larger key)
__device__ __forceinline__ unsigned fkey(float f) {
  unsigned u = __float_as_uint(f);
  return (u & 0x80000000u) ? ~u : (u | 0x80000000u);
}

// gate * base_rank_weight * |logit - mv|
__device__ __forceinline__ float rank_term(float lg, float m) {
  float l1 = fabsf(lg - m);
  float w = (m + 0.5f) * 0.5f;
  float gate = ((lg < m) || (l1 > 0.1f)) ? 1.0f : 0.0f;
  return gate * w * l1;
}

// ---------------- Pass 1: base rank-loss partial sums + coarse histogram ----
__global__ __launch_bounds__(TPB) void k1_scan(const float* __restrict__ logit,
                                               const float* __restrict__ mv,
                                               float* __restrict__ partial,
                                               unsigned* __restrict__ hist1,
                                               long n) {
  const long n4 = n >> 2;
  const long tid = (long)blockIdx.x * blockDim.x + threadIdx.x;
  const long stride = (long)gridDim.x * blockDim.x;
  const float4* lg4 = (const float4*)logit;
  const float4* mv4 = (const float4*)mv;
  float acc = 0.0f;
  for (long i = tid; i < n4; i += stride) {
    float4 lg = lg4[i];
    float4 m = mv4[i];
    __builtin_prefetch(lg4 + i + stride, 0, 1);
    __builtin_prefetch(mv4 + i + stride, 0, 1);
    acc += rank_term(lg.x, m.x);
    acc += rank_term(lg.y, m.y);
    acc += rank_term(lg.z, m.z);
    acc += rank_term(lg.w, m.w);
    atomicAdd(&hist1[fkey(m.x) >> 16], 1u);
    atomicAdd(&hist1[fkey(m.y) >> 16], 1u);
    atomicAdd(&hist1[fkey(m.z) >> 16], 1u);
    atomicAdd(&hist1[fkey(m.w) >> 16], 1u);
  }
  const long tail = n4 << 2;
  if (tail + tid < n) {
    float lg = logit[tail + tid];
    float m = mv[tail + tid];
    acc += rank_term(lg, m);
    atomicAdd(&hist1[fkey(m) >> 16], 1u);
  }
  __shared__ float red[TPB];
  red[threadIdx.x] = acc;
  __syncthreads();
  for (int s = TPB >> 1; s > 0; s >>= 1) {
    if ((int)threadIdx.x < s) red[threadIdx.x] += red[threadIdx.x + s];
    __syncthreads();
  }
  if (threadIdx.x == 0) partial[blockIdx.x] = red[0];
}

// ---------------- Pick coarse bin containing the K-th largest --------------
__global__ void k2_coarse(const unsigned* __restrict__ hist1,
                          unsigned* __restrict__ sel) {
  if (threadIdx.x == 0) {
    unsigned cum = 0;
    int b = 65535;
    for (; b >= 0; --b) {
      unsigned c = hist1[b];
      if (cum + c >= (unsigned)KTOP) break;
      cum += c;
    }
    if (b < 0) b = 0;
    sel[0] = (unsigned)b;  // coarse bin
    sel[1] = cum;          // count strictly above coarse bin
  }
}

// ---------------- Pass 2: fine histogram within the coarse bin -------------
__global__ __launch_bounds__(TPB) void k3_refine(const float* __restrict__ mv,
                                                 const unsigned* __restrict__ sel,
                                                 unsigned* __restrict__ hist2,
                                                 long n) {
  const unsigned coarse = sel[0];
  const long n4 = n >> 2;
  const long tid = (long)blockIdx.x * blockDim.x + threadIdx.x;
  const long stride = (long)gridDim.x * blockDim.x;
  const float4* mv4 = (const float4*)mv;
  for (long i = tid; i < n4; i += stride) {
    float4 m = mv4[i];
    unsigned k;
    k = fkey(m.x); if ((k >> 16) == coarse) atomicAdd(&hist2[k & 0xFFFFu], 1u);
    k = fkey(m.y); if ((k >> 16) == coarse) atomicAdd(&hist2[k & 0xFFFFu], 1u);
    k = fkey(m.z); if ((k >> 16) == coarse) atomicAdd(&hist2[k & 0xFFFFu], 1u);
    k = fkey(m.w); if ((k >> 16) == coarse) atomicAdd(&hist2[k & 0xFFFFu], 1u);
  }
  const long tail = n4 << 2;
  if (tail + tid < n) {
    unsigned k = fkey(mv[tail + tid]);
    if ((k >> 16) == coarse) atomicAdd(&hist2[k & 0xFFFFu], 1u);
  }
}

// ---------------- Exact 32-bit threshold key -------------------------------
__global__ void k4_fine(const unsigned* __restrict__ hist2,
                        unsigned* __restrict__ sel) {
  if (threadIdx.x == 0) {
    unsigned need = (unsigned)KTOP - sel[1];
    unsigned cum = 0;
    unsigned T = sel[0] << 16;
    for (int b = 65535; b >= 0; --b) {
      cum += hist2[b];
      if (cum >= need) { T = (sel[0] << 16) | (unsigned)b; break; }
    }
    sel[2] = T;  // threshold: collect key >= T
  }
}

// ---------------- Pass 3: compact candidates (key >= T) --------------------
__global__ __launch_bounds__(TPB) void k5_collect(const float* __restrict__ mv,
                                                  const unsigned* __restrict__ sel,
                                                  unsigned* __restrict__ ccnt,
                                                  unsigned* __restrict__ ckey,
                                                  unsigned* __restrict__ cidx,
                                                  long n) {
  const unsigned T = sel[2];
  const long n4 = n >> 2;
  const long tid = (long)blockIdx.x * blockDim.x + threadIdx.x;
  const long stride = (long)gridDim.x * blockDim.x;
  const float4* mv4 = (const float4*)mv;
  for (long i = tid; i < n4; i += stride) {
    float4 m = mv4[i];
    unsigned k;
#define PUSH(comp, off)                                                     \
    k = fkey(comp);                                                         \
    if (k >= T) {                                                           \
      unsigned p = atomicAdd(ccnt, 1u);                                     \
      if (p < CAND_MAX) { ckey[p] = k; cidx[p] = (unsigned)(4 * i + off); } \
    }
    PUSH(m.x, 0) PUSH(m.y, 1) PUSH(m.z, 2) PUSH(m.w, 3)
#undef PUSH
  }
  const long tail = n4 << 2;
  if (tail + tid < n) {
    unsigned k = fkey(mv[tail + tid]);
    if (k >= T) {
      unsigned p = atomicAdd(ccnt, 1u);
      if (p < CAND_MAX) { ckey[p] = k; cidx[p] = (unsigned)(tail + tid); }
    }
  }
}

// ---------------- Finalize: sort top-K, correction, WMMA gap loss ----------
__global__ __launch_bounds__(32) void k6_finalize(const float* __restrict__ logit,
                                                  const float* __restrict__ mv,
                                                  const unsigned* __restrict__ ckey,
                                                  const unsigned* __restrict__ cidx,
                                                  const unsigned* __restrict__ ccnt,
                                                  const float* __restrict__ partial,
                                                  int nblk, float* __restrict__ out,
                                                  long n) {
  __shared__ unsigned skey[CAND_MAX];
  __shared__ unsigned sidx[CAND_MAX];
  __shared__ float lt[64];
  __shared__ float rg[32], rc[32], rb[32];
  __shared__ int rn[32];
  const int t = threadIdx.x;
  const int m = min((int)*ccnt, CAND_MAX);

  for (int i = t; i < m; i += 32) { skey[i] = ckey[i]; sidx[i] = cidx[i]; }
  __syncthreads();

  // Deterministic selection of top-K: key desc, index asc (top_k tie rule)
  if (t == 0) {
    const int lim = min(KTOP, m);
    for (int r = 0; r < lim; ++r) {
      int best = r;
      for (int j = r + 1; j < m; ++j) {
        if (skey[j] > skey[best] ||
            (skey[j] == skey[best] && sidx[j] < sidx[best]))
          best = j;
      }
      unsigned tk = skey[r]; skey[r] = skey[best]; skey[best] = tk;
      unsigned ti = sidx[r]; sidx[r] = sidx[best]; sidx[best] = ti;
    }
  }
  __syncthreads();

  // Gather top-K logits (padded to 64 for 4x4 WMMA tiling)
  for (int r = t; r < 64; r += 32)
    lt[r] = (r < KTOP && r < m) ? logit[sidx[r]] : 0.0f;
  __syncthreads();

  // Rank-weight correction: add gate*w*l1*(factor-1) for top-K entries
  float corr = 0.0f;
  for (int r = t; r < KTOP; r += 32) {
    if (r < m) {
      unsigned i = sidx[r];
      float lg = logit[i], mm = mv[i];
      float q = (float)(KTOP - r) / (float)KTOP;
      float fac = 2.0f * (q * q * q * 4.0f + 1.0f);
      corr += rank_term(lg, mm) * (fac - 1.0f);
    }
  }

  // Deterministic reduction of pass-1 block partials
  float base = 0.0f;
  for (int j = t; j < nblk; j += 32) base += partial[j];

  // Pairwise gap loss via V_WMMA_F32_16X16X4_F32:
  //   D[mi,nj] = lt[mi]*1 + 1*(-lt[nj])  (rank-2 outer difference)
  // Per-lane operand staging (wave32):
  //   A 16x4: lanes 0-15 hold row M=lane {K0,K1}; lanes 16-31 hold {K2,K3}=0
  //   B 4x16: lanes 0-15 hold col N=lane {K0,K1}; lanes 16-31 hold {K2,K3}=0
  float gap = 0.0f;
  int num = 0;
  const float aval = (t < 16) ? 1.0f : 0.0f;  // A[:,1] / B[0,:] ones
#pragma unroll
  for (int tm = 0; tm < 4; ++tm) {
#pragma unroll
    for (int tn = 0; tn < 4; ++tn) {
      v2f a, b;
      a.x = (t < 16) ? lt[tm * 16 + t] : 0.0f;
      a.y = aval;
      b.x = aval;
      b.y = (t < 16) ? -lt[tn * 16 + t] : 0.0f;
      v8f c = {};
      c = __builtin_amdgcn_wmma_f32_16x16x4_f32(false, a, false, b,
                                                (short)0, c, false, false);
      // C/D layout: lane 0-15 -> N=lane, VGPR r -> M=r;
      //             lane 16-31 -> N=lane-16, VGPR r -> M=r+8
      const int nj = tn * 16 + (t & 15);
      const int mb = tm * 16 + ((t >> 4) * 8);
#pragma unroll
      for (int r = 0; r < 8; ++r) {
        const int mi = mb + r;
        const float d = c[r];
        // Branchless masked accumulation: pure per-lane predicate -> selects,
        // no EXEC save/restore chains around the WMMA results.
        const bool valid = (mi < KTOP) & (nj < KTOP) & (mi < nj) &
                           (fabsf(d) < 0.05f);
        const float w = valid ? 1.0f : 0.0f;
        gap += w * fmaxf(0.0f, 0.1f - d);
        num += valid ? 1 : 0;
      }
    }
  }

  rg[t] = gap; rc[t] = corr; rb[t] = base; rn[t] = num;
  __syncthreads();
  if (t == 0) {
    float G = 0.0f, C = 0.0f, B = 0.0f;
    int Nn = 0;
    for (int j = 0; j < 32; ++j) { G += rg[j]; C += rc[j]; B += rb[j]; Nn += rn[j]; }
    float gap_loss = G / fmaxf(1.0f, (float)Nn);
    out[0] = (B + C) / (float)n + gap_loss;
  }
}

extern "C" void kernel_launch(void* const* d_in, const int* in_sizes, int n_in,
                              void* d_out, int out_size, void* d_ws, size_t ws_size,
                              hipStream_t stream) {
  const float* logit = (const float*)d_in[0];
  const float* mv = (const float*)d_in[1];
  const long n = (long)in_sizes[0];

  unsigned* hist1 = (unsigned*)d_ws;            // 65536
  unsigned* hist2 = hist1 + 65536;              // 65536
  unsigned* sel = hist2 + 65536;                // 8
  unsigned* ccnt = sel + 8;                     // 8 (slot 0 used)
  unsigned* ckey = ccnt + 8;                    // CAND_MAX
  unsigned* cidx = ckey + CAND_MAX;             // CAND_MAX
  float* partial = (float*)(cidx + CAND_MAX);   // NB1
  float* out = (float*)d_out;

  hipMemsetAsync(hist1, 0, 2 * 65536 * sizeof(unsigned), stream);
  hipMemsetAsync(ccnt, 0, sizeof(unsigned), stream);

  k1_scan<<<NB1, TPB, 0, stream>>>(logit, mv, partial, hist1, n);
  k2_coarse<<<1, 32, 0, stream>>>(hist1, sel);
  k3_refine<<<NB1, TPB, 0, stream>>>(mv, sel, hist2, n);
  k4_fine<<<1, 32, 0, stream>>>(hist2, sel);
  k5_collect<<<NB1, TPB, 0, stream>>>(mv, sel, ccnt, ckey, cidx, n);
  k6_finalize<<<1, 32, 0, stream>>>(logit, mv, ckey, cidx, ccnt, partial, NB1, out, n);
}